// VQLookUpTable_26182120636831
// MI455X (gfx1250) — compile-verified
//
#include <hip/hip_runtime.h>
#include <hip/hip_bf16.h>

typedef __attribute__((ext_vector_type(2))) float v2f;
typedef __attribute__((ext_vector_type(8))) float v8f;

#define N_EMB     1024
#define DIM       256
#define N_VEC     32768           // 32*32*32 spatial vectors
#define SPATIAL   1024            // 32*32 per batch image
#define Q_ELEMS   8388608         // 32*256*32*32
#define LOSS_OFF  Q_ELEMS
#define USAGE_OFF (Q_ELEMS + 1)
#define LOSS_SCALE (1.25f / 8388608.0f)   // (1 + commitment_cost) / numel

// ---------------------------------------------------------------------------
// Init: zero loss + usage slots in d_out, precompute ||e||^2 per code into ws.
// Runs every call (graph replay re-accumulates, so re-zero is mandatory).
// ---------------------------------------------------------------------------
__global__ void vq_init_kernel(const float* __restrict__ w,
                               float* __restrict__ out,
                               float* __restrict__ wnorm) {
    int tid = blockIdx.x * 256 + threadIdx.x;   // 0..1023
    if (tid == 0) out[LOSS_OFF] = 0.0f;
    const float4* wr = (const float4*)(w + (size_t)tid * DIM);
    float s = 0.0f;
#pragma unroll
    for (int i = 0; i < DIM / 4; ++i) {
        float4 v = wr[i];
        s += v.x * v.x + v.y * v.y + v.z * v.z + v.w * v.w;
    }
    wnorm[tid] = s;
    out[USAGE_OFF + tid] = 0.0f;
}

// ---------------------------------------------------------------------------
// Fused distance-GEMM (V_WMMA_F32_16X16X4_F32) + argmin + gather/scatter +
// loss + usage histogram.  Double-buffered LDS codebook tiles: global loads
// for tile j+1 are issued before the 64-WMMA compute of tile j (latency
// hidden), ds_store into the alternate buffer after compute, one barrier per
// tile.
//   block = 256 threads (8 waves), each wave owns 16 input rows (M=16)
//   grid  = 32768 / 128 = 256 workgroups
// ---------------------------------------------------------------------------
__global__ void __launch_bounds__(256)
vq_main_kernel(const float* __restrict__ in,      // [32,256,32,32]
               const float* __restrict__ w,       // [1024,256]
               const float* __restrict__ wnorm,   // [1024] ||e||^2
               float* __restrict__ out) {
    // Two codebook tiles (16 codes x 256 K each), pair-interleaved [kpair][code]
    __shared__ v2f   ldsw[2][2048];  // 2 x 16 KB
    __shared__ int   idx_lds[128];   // chosen code per row of this WG
    __shared__ float loss_lds[8];

    const int tid  = threadIdx.x;
    const int lane = tid & 31;
    const int wave = tid >> 5;
    const int c    = lane & 15;      // N-column / M-row within 16-group
    const int hi   = lane >> 4;      // upper-half-wave flag (K phase)

    const int n0 = blockIdx.x * 128 + wave * 16;  // first row of this wave
    const int b  = n0 >> 10;                      // batch (16-row group never crosses)
    const int s0 = n0 & (SPATIAL - 1);            // spatial offset h*32+w

    // ---- Load A fragments: 16x256 slice, 64 k-steps of 16x4 f32 ----
    // ISA layout (32-bit A 16x4): lanes 0-15 -> K0(V0),K1(V1); lanes 16-31 -> K2,K3
    const float* xbase = in + (size_t)b * (DIM * SPATIAL) + (s0 + c);
    v2f a[64];
#pragma unroll
    for (int t = 0; t < 64; ++t) {
        const int kk = 4 * t + 2 * hi;
        v2f av;
        av.x = xbase[(size_t)kk * SPATIAL];
        av.y = xbase[(size_t)(kk + 1) * SPATIAL];
        a[t] = av;
    }

    // Running argmin of (||e||^2 - 2 x.e)  (||x||^2 dropped: constant per row)
    float vmin[8];
    int   vidx[8];
#pragma unroll
    for (int r = 0; r < 8; ++r) { vmin[r] = 3.4e38f; vidx[r] = 0; }

    // Cooperative-fill mapping: thread -> (code ci, k-pair chunk ii)
    const int ci = tid >> 4;   // 0..15 code within tile
    const int ii = tid & 15;
    v2f pref[8];

    // ---- Prologue: tile 0 -> LDS buffer 0 ----
    {
        const float* wp = w + (size_t)ci * DIM;
#pragma unroll
        for (int q = 0; q < 8; ++q) {
            const int p = ii + 16 * q;
            v2f wv; wv.x = wp[2 * p]; wv.y = wp[2 * p + 1];
            pref[q] = wv;
        }
#pragma unroll
        for (int q = 0; q < 8; ++q)
            ldsw[0][(ii + 16 * q) * 16 + ci] = pref[q];
    }
    __syncthreads();

    // ---- Stream the 64 codebook tiles (double-buffered, 1 barrier/tile) ----
    for (int j = 0; j < 64; ++j) {
        const int buf = j & 1;

        // Issue global loads for tile j+1 now; they retire under the WMMAs.
        if (j + 1 < 64) {
            const float* wp = w + (size_t)(16 * (j + 1) + ci) * DIM;
#pragma unroll
            for (int q = 0; q < 8; ++q) {
                const int p = ii + 16 * q;
                v2f wv; wv.x = wp[2 * p]; wv.y = wp[2 * p + 1];
                pref[q] = wv;
            }
        }

        // Compute 16x16 dot tile over K=256; B reads pipelined 1 iter ahead.
        const v2f* bufp = &ldsw[buf][0];
        v8f acc0 = {};
        v8f acc1 = {};
        v2f cb0 = bufp[(0 + hi) * 16 + c];
        v2f cb1 = bufp[(2 + hi) * 16 + c];
#pragma unroll
        for (int t = 0; t < 64; t += 2) {
            v2f nb0 = cb0, nb1 = cb1;
            if (t + 2 < 64) {
                nb0 = bufp[(2 * (t + 2) + hi) * 16 + c];
                nb1 = bufp[(2 * (t + 3) + hi) * 16 + c];
            }
            acc0 = __builtin_amdgcn_wmma_f32_16x16x4_f32(
                false, a[t], false, cb0, (short)0, acc0, false, false);
            acc1 = __builtin_amdgcn_wmma_f32_16x16x4_f32(
                false, a[t + 1], false, cb1, (short)0, acc1, false, false);
            cb0 = nb0;
            cb1 = nb1;
        }

        const float wn    = wnorm[16 * j + c];
        const int   ncode = 16 * j + c;
#pragma unroll
        for (int r = 0; r < 8; ++r) {
            const float score = wn - 2.0f * (acc0[r] + acc1[r]);
            if (score < vmin[r]) { vmin[r] = score; vidx[r] = ncode; }
        }

        // Park tile j+1 in the alternate buffer (loads have long landed).
        if (j + 1 < 64) {
#pragma unroll
            for (int q = 0; q < 8; ++q)
                ldsw[buf ^ 1][(ii + 16 * q) * 16 + ci] = pref[q];
        }
        __syncthreads();   // fill visible; all waves done reading ldsw[buf]
    }

    // ---- Cross-lane argmin within each 16-lane N-group (tie -> lowest idx) ----
#pragma unroll
    for (int off = 8; off >= 1; off >>= 1) {
#pragma unroll
        for (int r = 0; r < 8; ++r) {
            const float om = __shfl_xor(vmin[r], off, 32);
            const int   oi = __shfl_xor(vidx[r], off, 32);
            if (om < vmin[r] || (om == vmin[r] && oi < vidx[r])) {
                vmin[r] = om; vidx[r] = oi;
            }
        }
    }
    // C/D layout: VGPR r holds row M = r + 8*hi; publish per-row winners
    if (c == 0) {
#pragma unroll
        for (int r = 0; r < 8; ++r)
            idx_lds[wave * 16 + 8 * hi + r] = vidx[r];
    }
    __syncthreads();

    // ---- Usage histogram (float adds are exact for counts < 2^24) ----
    if (tid < 128)
        atomicAdd(&out[USAGE_OFF + idx_lds[tid]], 1.0f);

    // ---- Gather codes, scatter into NCHW, fused squared-error ----
    const int    code  = idx_lds[wave * 16 + c];        // this lane's row = n0 + c
    const float* wrow  = w + (size_t)code * DIM;
    const size_t obase = (size_t)b * (DIM * SPATIAL) + (s0 + c);
    float lacc = 0.0f;
#pragma unroll 4
    for (int d0 = 0; d0 < DIM; d0 += 2) {
        const int d = d0 + hi;                          // lanes 0-15: d0, 16-31: d0+1
        const float  v   = wrow[d];
        const size_t off = obase + (size_t)d * SPATIAL; // 64B-coalesced per half-wave
        const float  x   = in[off];
        const float  df  = v - x;
        lacc += df * df;
        out[off] = v;                                   // quantized_st == quantized
    }

    // ---- Loss reduction: wave shuffle -> LDS -> single atomic per WG ----
#pragma unroll
    for (int off = 16; off >= 1; off >>= 1)
        lacc += __shfl_xor(lacc, off, 32);
    if (lane == 0) loss_lds[wave] = lacc;
    __syncthreads();
    if (tid == 0) {
        float t = 0.0f;
#pragma unroll
        for (int ww = 0; ww < 8; ++ww) t += loss_lds[ww];
        atomicAdd(&out[LOSS_OFF], t * LOSS_SCALE);
    }
}

extern "C" void kernel_launch(void* const* d_in, const int* in_sizes, int n_in,
                              void* d_out, int out_size, void* d_ws, size_t ws_size,
                              hipStream_t stream) {
    const float* in = (const float*)d_in[0];   // [32,256,32,32] fp32
    const float* w  = (const float*)d_in[1];   // [1024,256]     fp32
    float* out   = (float*)d_out;              // quantized | loss | usage
    float* wnorm = (float*)d_ws;               // 1024 floats of scratch

    vq_init_kernel<<<4, 256, 0, stream>>>(w, out, wnorm);
    vq_main_kernel<<<N_VEC / 128, 256, 0, stream>>>(in, w, wnorm, out);
}